// EulerFormer_11184094839238
// MI455X (gfx1250) — compile-verified
//
#include <hip/hip_runtime.h>
#include <hip/hip_bf16.h>

typedef __attribute__((ext_vector_type(2))) float v2f;
typedef __attribute__((ext_vector_type(8))) float v8f;

#define BATCH 16
#define SEQ   4096
#define DIM   256
#define CHUNKS 32
#define ROWS_PER_CHUNK (SEQ / CHUNKS)   // 128
#define LDSTR 260                        // padded LDS row stride (floats): banks (4m+k)%64, conflict-free

// ---------------- Kernel 1: partial column sums of v over L ----------------
__global__ __launch_bounds__(256) void ef_partial_sums(const float* __restrict__ v,
                                                       float* __restrict__ partial) {
  const int c = blockIdx.x;      // chunk of L
  const int b = blockIdx.y;      // batch
  const int d = threadIdx.x;     // 0..255
  const float* p = v + ((size_t)b * SEQ + (size_t)c * ROWS_PER_CHUNK) * DIM + d;
  float acc = 0.f;
  #pragma unroll 4
  for (int l = 0; l < ROWS_PER_CHUNK; ++l) acc += p[(size_t)l * DIM];
  partial[((size_t)c * BATCH + b) * DIM + d] = acc;
}

// ---------------- Kernel 2: finish mean + Bsum ----------------
__global__ __launch_bounds__(256) void ef_scores_bsum(const float* __restrict__ partial,
                                                      float* __restrict__ s_out,
                                                      float* __restrict__ bsum_out) {
  const int b = blockIdx.x;
  const int i = threadIdx.x;
  float acc = 0.f;
  for (int c = 0; c < CHUNKS; ++c) acc += partial[((size_t)c * BATCH + b) * DIM + i];
  const float s = acc * (1.0f / SEQ);
  __shared__ float sl[DIM];
  sl[i] = s;
  __syncthreads();
  float bs = 0.f;
  for (int k = 0; k < DIM; ++k) bs += fabsf(s - sl[k]);
  s_out[b * DIM + i]    = s;
  bsum_out[b * DIM + i] = bs;
}

// ---------------- Kernel 3: stable softmax rows of P ----------------
// P[b][j][i] = softmax_i( s_i * (255 - 2j) - Bsum_i )   (tau == 1)
__global__ __launch_bounds__(256) void ef_softmax(const float* __restrict__ s_in,
                                                  const float* __restrict__ bsum_in,
                                                  float* __restrict__ P) {
  const int j = blockIdx.x;
  const int b = blockIdx.y;
  const int i = threadIdx.x;
  const float s  = s_in[b * DIM + i];
  const float bs = bsum_in[b * DIM + i];
  const float scal = (float)(DIM - 1 - 2 * j);
  const float x = s * scal - bs;
  __shared__ float red[DIM];
  red[i] = x; __syncthreads();
  for (int t = DIM / 2; t > 0; t >>= 1) {
    if (i < t) red[i] = fmaxf(red[i], red[i + t]);
    __syncthreads();
  }
  const float m = red[0];
  __syncthreads();
  const float e = expf(x - m);
  red[i] = e; __syncthreads();
  for (int t = DIM / 2; t > 0; t >>= 1) {
    if (i < t) red[i] += red[i + t];
    __syncthreads();
  }
  P[((size_t)b * DIM + j) * DIM + i] = e / red[0];
}

// ---------------- Kernel 4: fused WMMA GEMM + Euler epilogue ----------------
// C[l,d] = sum_e v[l,e] * P[d,e].  Block tile 64(M) x 64(N), 8 waves (2 M-halves
// x 4 N-tiles); each wave computes TWO 16x16 tiles stacked in M, sharing one
// B fragment per k-step (3 DS loads : 2 WMMAs).
__global__ __launch_bounds__(256) void ef_gemm_euler(const float* __restrict__ v,
                                                     const float* __restrict__ P,
                                                     const float* __restrict__ delta,
                                                     const float* __restrict__ log_scale,
                                                     float* __restrict__ out) {
  extern __shared__ float lds[];
  float* lds_a = lds;               // 64 x LDSTR  (v tile)
  float* lds_b = lds + 64 * LDSTR;  // 64 x LDSTR  (P rows = B^T tile)

  const int tid  = threadIdx.x;
  const int lane = tid & 31;
  const int wave = tid >> 5;
  const int mw = wave & 1;          // M half: rows [mw*32, mw*32+32)
  const int nw = wave >> 1;         // N tile: cols [nw*16, nw*16+16)

  const int l_base = blockIdx.x * 64;   // 64 tiles over L
  const int n_base = blockIdx.y * 64;   // 4 tiles over D
  const int b      = blockIdx.z;        // 16 batches

  // stage A tile: 64 rows x 256 cols of v (4096 float4, coalesced)
  {
    const float4* src = (const float4*)(v + ((size_t)b * SEQ + l_base) * DIM);
    #pragma unroll
    for (int it = 0; it < 16; ++it) {
      const int idx = tid + it * 256;
      const int row = idx >> 6;
      const int c4  = idx & 63;
      const float4 val = src[(size_t)row * 64 + c4];
      float* dst = &lds_a[row * LDSTR + c4 * 4];
      dst[0] = val.x; dst[1] = val.y; dst[2] = val.z; dst[3] = val.w;
    }
  }
  // stage B tile: P rows n_base..n_base+63, all 256 k (4096 float4, coalesced)
  {
    const float4* src = (const float4*)(P + ((size_t)b * DIM + n_base) * DIM);
    #pragma unroll
    for (int it = 0; it < 16; ++it) {
      const int idx = tid + it * 256;
      const int row = idx >> 6;
      const int c4  = idx & 63;
      const float4 val = src[(size_t)row * 64 + c4];
      float* dst = &lds_b[row * LDSTR + c4 * 4];
      dst[0] = val.x; dst[1] = val.y; dst[2] = val.z; dst[3] = val.w;
    }
  }
  __syncthreads();

  // fragment addressing per ISA 16x4 f32 A layout:
  // lane<16: M=lane, K={k0,k0+1}; lane>=16: M=lane-16, K={k0+2,k0+3}
  const int mn = lane & 15;
  const int kk = (lane >> 4) << 1;
  const float* ap0 = &lds_a[(mw * 32 + mn) * LDSTR + kk];        // M tile 0
  const float* ap1 = ap0 + 16 * LDSTR;                            // M tile 1
  const float* bp  = &lds_b[(nw * 16 + mn) * LDSTR + kk];

  v8f acc0 = {};
  v8f acc1 = {};
  #pragma unroll
  for (int k0 = 0; k0 < DIM; k0 += 4) {
    const v2f a0 = *(const v2f*)(ap0 + k0);
    const v2f a1 = *(const v2f*)(ap1 + k0);
    const v2f bb = *(const v2f*)(bp + k0);
    acc0 = __builtin_amdgcn_wmma_f32_16x16x4_f32(false, a0, false, bb,
                                                 (short)0, acc0, false, false);
    acc1 = __builtin_amdgcn_wmma_f32_16x16x4_f32(false, a1, false, bb,
                                                 (short)0, acc1, false, false);
  }

  // Euler epilogue: pair columns (n even -> r, n odd -> p)
  const int  n_g  = n_base + nw * 16 + mn;
  const int  h    = n_g >> 1;
  const bool even = (mn & 1) == 0;
  const float dl  = delta[h];
  float ls = log_scale[h];
  ls = fminf(fmaxf(ls, -5.0f), 5.0f);
  const float esc = __expf(ls);
  const int mhi = (lane < 16) ? 0 : 8;

  #pragma unroll
  for (int t = 0; t < 2; ++t) {
    const v8f acc = t ? acc1 : acc0;
    const int l_t = l_base + mw * 32 + t * 16 + mhi;
    #pragma unroll
    for (int i = 0; i < 8; ++i) {
      const float cval  = acc[i];
      const float other = __shfl_xor(cval, 1, 32);  // partner column (n ^ 1), same row
      const float r = even ? cval  : other;
      const float p = even ? other : cval;
      const float lam  = sqrtf(r * r + p * p + 1e-6f);
      const float th   = atan2f(p, r) * dl;         // atan2(p/lam, r/lam) == atan2(p, r)
      const float lam2 = lam * esc;
      const float o    = even ? (lam2 * __cosf(th)) : (lam2 * __sinf(th));
      out[((size_t)b * SEQ + (l_t + i)) * DIM + n_g] = o;
    }
  }
}

extern "C" void kernel_launch(void* const* d_in, const int* in_sizes, int n_in,
                              void* d_out, int out_size, void* d_ws, size_t ws_size,
                              hipStream_t stream) {
  const float* v         = (const float*)d_in[0];
  const float* delta     = (const float*)d_in[1];
  // d_in[2] (b) is unused by the reference computation
  const float* log_scale = (const float*)d_in[3];
  float* out = (float*)d_out;

  // scratch layout (floats): partial sums | s | Bsum | P   (~4.53 MB total)
  float* ws      = (float*)d_ws;
  float* partial = ws;                                       // 32*16*256
  float* s_ws    = partial + (size_t)CHUNKS * BATCH * DIM;   // 16*256
  float* bsum_ws = s_ws + (size_t)BATCH * DIM;               // 16*256
  float* P_ws    = bsum_ws + (size_t)BATCH * DIM;            // 16*256*256

  ef_partial_sums<<<dim3(CHUNKS, BATCH), 256, 0, stream>>>(v, partial);
  ef_scores_bsum <<<dim3(BATCH),          256, 0, stream>>>(partial, s_ws, bsum_ws);
  ef_softmax     <<<dim3(DIM, BATCH),     256, 0, stream>>>(s_ws, bsum_ws, P_ws);

  const size_t shmem = (size_t)(2 * 64 * LDSTR) * sizeof(float); // 133120 B
  ef_gemm_euler<<<dim3(SEQ / 64, DIM / 64, BATCH), 256, shmem, stream>>>(
      v, P_ws, delta, log_scale, out);
}